// mIoUEstimator_44470091383038
// MI455X (gfx1250) — compile-verified
//
#include <hip/hip_runtime.h>

// mIoU estimator for (64,512,512) float32 class-id preds + int32 targets, 91 classes.
// Bandwidth-bound: 134 MB read -> ~5.8us floor at 23.3 TB/s.
// Stage 1: per-wave privatized LDS histograms (ds_add_u32) + global atomic merge.
// Stage 2: single-wave IoU reduction using v_wmma_f32_16x16x4_f32 for the
//          32-lane cross-lane sums (CDNA5 matrix path).

#define N_CLASSES 91
#define NBINS 96          // padded to 96 (multiple of 32, avoids edge cases)
#define WPB 8             // waves per block (256 / 32)
#define BLOCK 256
#define NB 1024           // grid size for the histogram kernel

typedef __attribute__((ext_vector_type(2))) float v2f;
typedef __attribute__((ext_vector_type(8))) float v8f;

__global__ void miou_init_bins(unsigned int* __restrict__ bins) {
  int i = blockIdx.x * blockDim.x + threadIdx.x;
  if (i < 3 * NBINS) bins[i] = 0u;
}

__global__ __launch_bounds__(BLOCK) void miou_hist(
    const float* __restrict__ x, const int* __restrict__ t,
    unsigned int* __restrict__ gbins, int n) {
  __shared__ unsigned int h[WPB * 3 * NBINS];  // 9216 B
  for (int i = threadIdx.x; i < WPB * 3 * NBINS; i += BLOCK) h[i] = 0u;
  __syncthreads();

  // Per-wave private histograms: no inter-wave LDS atomic contention.
  unsigned int* ha = &h[(threadIdx.x >> 5) * 3 * NBINS];
  unsigned int* hb = ha + NBINS;       // target counts
  unsigned int* hi = hb + NBINS;       // intersection counts

  const int tid    = blockIdx.x * BLOCK + threadIdx.x;
  const int stride = BLOCK * NB;
  const int n4     = n >> 2;

  const float4* __restrict__ x4 = (const float4*)x;
  const int4*   __restrict__ t4 = (const int4*)t;

  for (int i = tid; i < n4; i += stride) {
    const float4 xv = x4[i];   // 128-bit coalesced loads
    const int4   tv = t4[i];
    int a;
    a = (int)xv.x; atomicAdd(&ha[a], 1u); atomicAdd(&hb[tv.x], 1u); if (a == tv.x) atomicAdd(&hi[a], 1u);
    a = (int)xv.y; atomicAdd(&ha[a], 1u); atomicAdd(&hb[tv.y], 1u); if (a == tv.y) atomicAdd(&hi[a], 1u);
    a = (int)xv.z; atomicAdd(&ha[a], 1u); atomicAdd(&hb[tv.z], 1u); if (a == tv.z) atomicAdd(&hi[a], 1u);
    a = (int)xv.w; atomicAdd(&ha[a], 1u); atomicAdd(&hb[tv.w], 1u); if (a == tv.w) atomicAdd(&hi[a], 1u);
  }
  // Scalar tail (n not divisible by 4).
  for (int i = (n4 << 2) + tid; i < n; i += stride) {
    int a = (int)x[i]; int b = t[i];
    atomicAdd(&ha[a], 1u); atomicAdd(&hb[b], 1u); if (a == b) atomicAdd(&hi[a], 1u);
  }
  __syncthreads();

  // Reduce the 8 wave-private copies and merge into global bins.
  for (int i = threadIdx.x; i < 3 * NBINS; i += BLOCK) {
    unsigned int s = 0;
#pragma unroll
    for (int w = 0; w < WPB; ++w) s += h[w * 3 * NBINS + i];
    if (s) atomicAdd(&gbins[i], s);
  }
}

// One wave32. EXEC is all-ones at the WMMA (divergence above reconverges).
__global__ void miou_finalize(const unsigned int* __restrict__ bins,
                              const int* __restrict__ smooth_p,
                              float* __restrict__ out) {
  const int lane = threadIdx.x;  // 0..31
  const float smooth = (float)smooth_p[0];

  float s_iou = 0.0f, n_valid = 0.0f;
  for (int c = lane; c < N_CLASSES; c += 32) {
    float a  = (float)bins[c];
    float b  = (float)bins[NBINS + c];
    float it = (float)bins[2 * NBINS + c];
    float un = a + b - it;
    if (un != 0.0f) {
      s_iou   += (it + smooth) / (un + smooth);
      n_valid += 1.0f;
    }
  }

  // Cross-lane reduction via V_WMMA_F32_16X16X4_F32.
  // A (16x4 f32): lane m holds A[m][0] (VGPR0) / A[m][1] (VGPR1) for lanes 0-15,
  // A[m][2]/A[m][3] for lanes 16-31. Put the partial in K=0/K=2 slot, 0 in the other.
  // B = all ones (layout-agnostic) => D[m][n] = partial(lane m) + partial(lane m+16).
  v2f A, B;
  v8f C = {};
  B.x = 1.0f; B.y = 1.0f;

  A.x = s_iou;   A.y = 0.0f;
  v8f Ds = __builtin_amdgcn_wmma_f32_16x16x4_f32(false, A, false, B,
                                                 (short)0, C, false, false);
  A.x = n_valid; A.y = 0.0f;
  v8f Dv = __builtin_amdgcn_wmma_f32_16x16x4_f32(false, A, false, B,
                                                 (short)0, C, false, false);

  // Per lane: sum the 8 D VGPRs = sum of rows {0..7} (lanes 0-15) or {8..15}
  // (lanes 16-31) of one column. lane0 + lane16 covers all 16 rows = full sum.
  float cs = Ds[0] + Ds[1] + Ds[2] + Ds[3] + Ds[4] + Ds[5] + Ds[6] + Ds[7];
  float cv = Dv[0] + Dv[1] + Dv[2] + Dv[3] + Dv[4] + Dv[5] + Dv[6] + Dv[7];

  float tot_s = __shfl(cs, 0, 32) + __shfl(cs, 16, 32);
  float tot_v = __shfl(cv, 0, 32) + __shfl(cv, 16, 32);

  if (lane == 0) out[0] = tot_s / tot_v;
}

extern "C" void kernel_launch(void* const* d_in, const int* in_sizes, int n_in,
                              void* d_out, int out_size, void* d_ws, size_t ws_size,
                              hipStream_t stream) {
  const float* x      = (const float*)d_in[0];   // inputs (float32 class ids)
  const int*   t      = (const int*)d_in[1];     // targets (int32 class ids)
  const int*   smooth = (const int*)d_in[2];     // scalar smooth
  float*       out    = (float*)d_out;
  unsigned int* bins  = (unsigned int*)d_ws;     // 3 * 96 u32
  const int n = in_sizes[0];

  miou_init_bins<<<1, 3 * NBINS, 0, stream>>>(bins);
  miou_hist<<<NB, BLOCK, 0, stream>>>(x, t, bins, n);
  miou_finalize<<<1, 32, 0, stream>>>(bins, smooth, out);
}